// LocalGeometryAggregation_3006477107872
// MI455X (gfx1250) — compile-verified
//
#include <hip/hip_runtime.h>
#include <math.h>

// Problem constants (match reference)
#define B_   8
#define N_   4096
#define CIN_ 128
#define D_   128
#define DH_  64        // D/2
#define K_   16
#define F2_  256       // 2*D (fusion input width)
#define EPS_ 1e-5f

typedef float v2f __attribute__((ext_vector_type(2)));
typedef float v8f __attribute__((ext_vector_type(8)));

// ---------------------------------------------------------------------------
// CDNA5 WMMA: D(16x16,f32) = A(16x4,f32) * B(4x16,f32) + C
// A frag (2 VGPR): lanes 0-15 row M=lane  k={0,1}; lanes 16-31 row M=lane-16 k={2,3}
// B frag (2 VGPR): lanes 0-15 col N=lane  k={0,1}? -> mirrored split: x holds
//                  k_local = 2*(lane>>4), y holds k_local+1 (same pattern as A)
// C/D (8 VGPR):    VGPR j: lanes 0-15 -> (M=j, N=lane); lanes 16-31 -> (M=j+8)
// ---------------------------------------------------------------------------
__device__ __forceinline__ v8f wmma4(v2f a, v2f b, v8f c) {
  return __builtin_amdgcn_wmma_f32_16x16x4_f32(false, a, false, b, (short)0, c,
                                               false, false);
}

__device__ __forceinline__ v8f v8f_zero() {
  v8f z = {0.f, 0.f, 0.f, 0.f, 0.f, 0.f, 0.f, 0.f};
  return z;
}

__device__ __forceinline__ float silu_f(float x) { return x / (1.f + expf(-x)); }

// ===========================================================================
// Kernel 1: exact KNN (top-16 smallest squared distances, self included).
// One block = 256 query points of one batch; full batch xyz cached in LDS as
// float4 (x,y,z,|x|^2).  Register-resident sorted insertion list (static
// indices only -> stays in VGPRs, no scratch).
// ===========================================================================
__global__ __launch_bounds__(256)
void knn_kernel(const float* __restrict__ xyz, int* __restrict__ idx)
{
  extern __shared__ float4 spts[];                 // 4096 * 16B = 64 KB
  const int b  = blockIdx.x >> 4;                  // 16 row-tiles per batch
  const int qn = ((blockIdx.x & 15) << 8) + threadIdx.x;
  const float* bx = xyz + (size_t)b * N_ * 3;

  for (int i = threadIdx.x; i < N_; i += 256) {
    float x = bx[i * 3 + 0], y = bx[i * 3 + 1], z = bx[i * 3 + 2];
    spts[i] = make_float4(x, y, z, x * x + y * y + z * z);
  }
  __syncthreads();

  const float4 q = spts[qn];
  float bd[K_]; int bi[K_];
#pragma unroll
  for (int t = 0; t < K_; ++t) { bd[t] = 3.4e38f; bi[t] = 0; }

  for (int j = 0; j < N_; ++j) {
    float4 pv = spts[j];
    float d = q.w + pv.w - 2.f * (q.x * pv.x + q.y * pv.y + q.z * pv.z);
    if (d < bd[K_ - 1]) {                          // rare: ~16*ln(N/K) inserts
      float cd = d; int ci = j;
#pragma unroll
      for (int t = 0; t < K_; ++t) {               // bubble-insert, keeps sort
        if (cd < bd[t]) {                          // strict < : stable ties
          float td = bd[t]; int ti = bi[t];
          bd[t] = cd; bi[t] = ci; cd = td; ci = ti;
        }
      }
    }
  }
  int* op = idx + ((size_t)b * N_ + qn) * K_;
#pragma unroll
  for (int t = 0; t < K_; ++t) op[t] = bi[t];
}

// ===========================================================================
// Kernel 2: global first/second moments of rel = nb_xyz - xyz  (9 sums).
// Deterministic: per-block LDS tree reduce -> partials; no float atomics.
// ===========================================================================
__global__ __launch_bounds__(256)
void rel_stats_kernel(const float* __restrict__ xyz, const int* __restrict__ idx,
                      float* __restrict__ part)
{
  __shared__ float red[256];
  const int p = blockIdx.x * 256 + threadIdx.x;    // 0..B*N-1
  const int b = p >> 12, n = p & (N_ - 1);
  const float* bx = xyz + (size_t)b * N_ * 3;
  const float cx = bx[n * 3 + 0], cy = bx[n * 3 + 1], cz = bx[n * 3 + 2];

  float s[9];
#pragma unroll
  for (int t = 0; t < 9; ++t) s[t] = 0.f;

  const int* ip = idx + (size_t)p * K_;
  for (int k = 0; k < K_; ++k) {
    int j = ip[k];
    float rx = bx[j * 3 + 0] - cx, ry = bx[j * 3 + 1] - cy, rz = bx[j * 3 + 2] - cz;
    s[0] += rx; s[1] += ry; s[2] += rz;
    s[3] += rx * rx; s[4] += rx * ry; s[5] += rx * rz;
    s[6] += ry * ry; s[7] += ry * rz; s[8] += rz * rz;
  }
  for (int qd = 0; qd < 9; ++qd) {
    red[threadIdx.x] = s[qd];
    __syncthreads();
    for (int st = 128; st > 0; st >>= 1) {
      if (threadIdx.x < st) red[threadIdx.x] += red[threadIdx.x + st];
      __syncthreads();
    }
    if (threadIdx.x == 0) part[blockIdx.x * 12 + qd] = red[0];
    __syncthreads();
  }
}

// ===========================================================================
// Kernel 3: fold moments through the (linear) conv1 analytically to get the
// exact train-mode BatchNorm batch stats; emit per-channel scale/shift.
//   h = w.r + c  =>  E[h] = w.m + c ;  Var[h] = w^T S2 w - (w.m)^2
// ===========================================================================
__global__ __launch_bounds__(64)
void bn_finalize_kernel(const float* __restrict__ part,
                        const float* __restrict__ cw1, const float* __restrict__ cb1,
                        const float* __restrict__ bn_g, const float* __restrict__ bn_b,
                        float* __restrict__ bn_st)        // [64] scale | [64] shift
{
  __shared__ float tot[9];
  if (threadIdx.x < 9) {
    float a = 0.f;
    for (int blk = 0; blk < (B_ * N_) / 256; ++blk) a += part[blk * 12 + threadIdx.x];
    tot[threadIdx.x] = a;
  }
  __syncthreads();

  const float inv = 1.f / (float)((size_t)B_ * N_ * K_);
  const int o = threadIdx.x;
  if (o < DH_) {
    float w0 = cw1[o * 3], w1 = cw1[o * 3 + 1], w2 = cw1[o * 3 + 2], c = cb1[o];
    float m0 = tot[0] * inv, m1 = tot[1] * inv, m2 = tot[2] * inv;
    float wm = w0 * m0 + w1 * m1 + w2 * m2;
    float e2 = (w0 * w0 * tot[3] + w1 * w1 * tot[6] + w2 * w2 * tot[8]
             + 2.f * (w0 * w1 * tot[4] + w0 * w2 * tot[5] + w1 * w2 * tot[7])) * inv;
    float var = e2 - wm * wm;
    if (var < 0.f) var = 0.f;
    float sc = bn_g[o] * rsqrtf(var + EPS_);
    bn_st[o]       = sc;
    bn_st[DH_ + o] = bn_b[o] - (wm + c) * sc;
  }
}

// ===========================================================================
// Kernel 4: fused per-point pipeline.  One wave32 per point; M=16 neighbor
// rows map directly onto the WMMA 16-row A operand.  tf / geo / fusion GEMMs
// all run on V_WMMA_F32_16X16X4_F32; tf+geo staged through a per-wave LDS
// panel XF[16][256] that is re-read as the fusion A operand.
// Per-wave LDS: 16*256 (XF) + 16*64 (geo hidden) + 32 (scratch) floats.
// ===========================================================================
#define WAVE_LDS (16 * F2_ + 16 * DH_ + 32)   // 5152 floats
#define WPB 4                                  // waves per block (128 threads)

__global__ __launch_bounds__(WPB * 32)
void lga_main_kernel(const float* __restrict__ xyz,  const float* __restrict__ feat,
                     const int*   __restrict__ idx,
                     const float* __restrict__ W1,   const float* __restrict__ b1,
                     const float* __restrict__ ln1g, const float* __restrict__ ln1b,
                     const float* __restrict__ cw1,  const float* __restrict__ cb1,
                     const float* __restrict__ bn_st,
                     const float* __restrict__ cw2,  const float* __restrict__ cb2,
                     const float* __restrict__ Wf,   const float* __restrict__ bf,
                     const float* __restrict__ lnfg, const float* __restrict__ lnfb,
                     const float* __restrict__ alpha,const float* __restrict__ beta,
                     float* __restrict__ out)
{
  extern __shared__ float sm[];
  const int lane = threadIdx.x & 31;
  const int wave = threadIdx.x >> 5;
  const int p    = blockIdx.x * WPB + wave;        // global point id (exact fit)
  const int b    = p >> 12;
  const int n    = p & (N_ - 1);
  const int m    = lane & 15;                      // matrix row (neighbor k)
  const int hi   = lane >> 4;                      // lane half
  const int kx   = hi << 1;                        // WMMA k sub-offset {0,2}

  float* XF = sm + wave * WAVE_LDS;                // [16][256] concat panel
  float* Hh = XF + 16 * F2_;                       // [16][64] geo hidden
  float* SC = Hh + 16 * DH_;                       // row-sum scratch

  // ---- neighbor gather + relative position for row m (both lane halves) ----
  const int nb = idx[(size_t)p * K_ + m];
  const float* bxyz = xyz + (size_t)b * N_ * 3;
  const float rx = bxyz[nb * 3 + 0] - bxyz[n * 3 + 0];
  const float ry = bxyz[nb * 3 + 1] - bxyz[n * 3 + 1];
  const float rz = bxyz[nb * 3 + 2] - bxyz[n * 3 + 2];

  // ---- geo hidden: h = SiLU(BN(rel @ cw1^T + cb1)), lane covers 32 channels
  for (int t = 0; t < 32; ++t) {
    int o = (hi << 5) + t;
    float h = rx * cw1[o * 3] + ry * cw1[o * 3 + 1] + rz * cw1[o * 3 + 2] + cb1[o];
    h = h * bn_st[o] + bn_st[DH_ + o];
    Hh[m * DH_ + o] = silu_f(h);
  }

  // ---- tf = nb_feat[16,128] @ W1[128,128]  (WMMA, 32 k-steps x 8 n-tiles) ----
  v8f acc[8];
#pragma unroll
  for (int nt = 0; nt < 8; ++nt) acc[nt] = v8f_zero();
  const float* arow = feat + ((size_t)b * N_ + nb) * CIN_;   // row m's features
  for (int k = 0; k < CIN_; k += 4) {
    v2f a; a.x = arow[k + kx]; a.y = arow[k + kx + 1];
#pragma unroll
    for (int nt = 0; nt < 8; ++nt) {
      const float* wp = W1 + (size_t)(k + kx) * D_ + (nt << 4) + m;
      v2f bb; bb.x = wp[0]; bb.y = wp[D_];
      acc[nt] = wmma4(a, bb, acc[nt]);
    }
  }
#pragma unroll
  for (int nt = 0; nt < 8; ++nt) {                 // +b1, stage to XF[:,0:128]
    int col = (nt << 4) + m;
    float bv = b1[col];
#pragma unroll
    for (int j = 0; j < 8; ++j)
      XF[(j + (hi << 3)) * F2_ + col] = acc[nt][j] + bv;
  }
  __syncthreads();

  // ---- tf LayerNorm + SiLU (one lane per row, two-pass) ----
  if (lane < 16) {
    float* row = XF + lane * F2_;
    float s = 0.f;
    for (int c = 0; c < D_; ++c) s += row[c];
    float mean = s * (1.f / 128.f);
    float v = 0.f;
    for (int c = 0; c < D_; ++c) { float d = row[c] - mean; v += d * d; }
    float ivs = rsqrtf(v * (1.f / 128.f) + EPS_);
    for (int c = 0; c < D_; ++c)
      row[c] = silu_f((row[c] - mean) * ivs * ln1g[c] + ln1b[c]);
  }
  __syncthreads();

  // ---- geo = h[16,64] @ cw2^T[64,128]  (WMMA, 16 k-steps x 8 n-tiles) ----
#pragma unroll
  for (int nt = 0; nt < 8; ++nt) acc[nt] = v8f_zero();
  for (int k = 0; k < DH_; k += 4) {
    v2f a; a.x = Hh[m * DH_ + k + kx]; a.y = Hh[m * DH_ + k + kx + 1];
#pragma unroll
    for (int nt = 0; nt < 8; ++nt) {
      const float* wp = cw2 + (size_t)((nt << 4) + m) * DH_ + k + kx;  // B=cw2^T
      v2f bb; bb.x = wp[0]; bb.y = wp[1];
      acc[nt] = wmma4(a, bb, acc[nt]);
    }
  }
#pragma unroll
  for (int nt = 0; nt < 8; ++nt) {                 // +cb2, stage to XF[:,128:256]
    int col = (nt << 4) + m;
    float bv = cb2[col];
#pragma unroll
    for (int j = 0; j < 8; ++j)
      XF[(j + (hi << 3)) * F2_ + D_ + col] = acc[nt][j] + bv;
  }
  __syncthreads();

  // ---- fused = XF[16,256] @ Wf[256,128]  (WMMA, 64 k-steps x 8 n-tiles) ----
#pragma unroll
  for (int nt = 0; nt < 8; ++nt) acc[nt] = v8f_zero();
  for (int k = 0; k < F2_; k += 4) {
    v2f a; a.x = XF[m * F2_ + k + kx]; a.y = XF[m * F2_ + k + kx + 1];
#pragma unroll
    for (int nt = 0; nt < 8; ++nt) {
      const float* wp = Wf + (size_t)(k + kx) * D_ + (nt << 4) + m;
      v2f bb; bb.x = wp[0]; bb.y = wp[D_];
      acc[nt] = wmma4(a, bb, acc[nt]);
    }
  }
  __syncthreads();                                 // all reads of XF done
#pragma unroll
  for (int nt = 0; nt < 8; ++nt) {                 // +bf, overwrite XF[:,0:128]
    int col = (nt << 4) + m;
    float bv = bf[col];
#pragma unroll
    for (int j = 0; j < 8; ++j)
      XF[(j + (hi << 3)) * F2_ + col] = acc[nt][j] + bv;
  }
  __syncthreads();

  // ---- fusion LN + SiLU + affine, accumulate row sums for softmax ----
  if (lane < 16) {
    float* row = XF + lane * F2_;
    float s = 0.f;
    for (int c = 0; c < D_; ++c) s += row[c];
    float mean = s * (1.f / 128.f);
    float v = 0.f;
    for (int c = 0; c < D_; ++c) { float d = row[c] - mean; v += d * d; }
    float ivs = rsqrtf(v * (1.f / 128.f) + EPS_);
    float rs = 0.f;
    for (int c = 0; c < D_; ++c) {
      float val = silu_f((row[c] - mean) * ivs * lnfg[c] + lnfb[c]);
      val = alpha[c] * val + beta[c];
      row[c] = val; rs += val;
    }
    SC[lane] = rs;
  }
  __syncthreads();

  // ---- softmax over K=16 + weighted aggregation (coalesced store) ----
  float mx = -3.4e38f;
#pragma unroll
  for (int t = 0; t < K_; ++t) mx = fmaxf(mx, SC[t]);
  float z = 0.f; float wgt[K_];
#pragma unroll
  for (int t = 0; t < K_; ++t) { wgt[t] = expf(SC[t] - mx); z += wgt[t]; }
  const float zi = 1.f / z;
  float* op = out + (size_t)p * D_;
#pragma unroll
  for (int t4 = 0; t4 < 4; ++t4) {
    int c = lane + (t4 << 5);
    float o = 0.f;
#pragma unroll
    for (int t = 0; t < K_; ++t) o += wgt[t] * XF[t * F2_ + c];
    op[c] = o * zi;
  }
}

// ===========================================================================
// Launcher — all kernels on `stream`, scratch layout in d_ws:
//   [0)            B*N*K ints   : KNN indices            (2 MB)
//   [+524288 f)    128*12 float : rel-stat partials
//   [+1536 f)      128 float    : BN scale|shift
// ===========================================================================
extern "C" void kernel_launch(void* const* d_in, const int* in_sizes, int n_in,
                              void* d_out, int out_size, void* d_ws, size_t ws_size,
                              hipStream_t stream)
{
  (void)in_sizes; (void)n_in; (void)out_size; (void)ws_size;
  const float* xyz  = (const float*)d_in[0];
  const float* feat = (const float*)d_in[1];
  const float* W1   = (const float*)d_in[2];
  const float* b1   = (const float*)d_in[3];
  const float* ln1g = (const float*)d_in[4];
  const float* ln1b = (const float*)d_in[5];
  const float* cw1  = (const float*)d_in[6];
  const float* cb1  = (const float*)d_in[7];
  const float* bng  = (const float*)d_in[8];
  const float* bnb  = (const float*)d_in[9];
  const float* cw2  = (const float*)d_in[10];
  const float* cb2  = (const float*)d_in[11];
  const float* Wf   = (const float*)d_in[12];
  const float* bf   = (const float*)d_in[13];
  const float* lnfg = (const float*)d_in[14];
  const float* lnfb = (const float*)d_in[15];
  const float* alph = (const float*)d_in[16];
  const float* beta = (const float*)d_in[17];
  float* out = (float*)d_out;

  int*   d_idx  = (int*)d_ws;
  float* d_part = (float*)d_ws + (size_t)B_ * N_ * K_;
  float* d_bn   = d_part + 128 * 12;

  knn_kernel<<<dim3(B_ * (N_ / 256)), 256, N_ * sizeof(float4), stream>>>(xyz, d_idx);
  rel_stats_kernel<<<dim3((B_ * N_) / 256), 256, 0, stream>>>(xyz, d_idx, d_part);
  bn_finalize_kernel<<<dim3(1), 64, 0, stream>>>(d_part, cw1, cb1, bng, bnb, d_bn);
  lga_main_kernel<<<dim3((B_ * N_) / WPB), WPB * 32,
                    WPB * WAVE_LDS * sizeof(float), stream>>>(
      xyz, feat, d_idx, W1, b1, ln1g, ln1b, cw1, cb1, d_bn, cw2, cb2,
      Wf, bf, lnfg, lnfb, alph, beta, out);
}